// MixingAttention_10411000725986
// MI455X (gfx1250) — compile-verified
//
#include <hip/hip_runtime.h>
#include <math.h>

typedef float v2f __attribute__((ext_vector_type(2)));
typedef float v8f __attribute__((ext_vector_type(8)));

#define EPSN 1e-5f

enum { MODE_PLAIN = 0, MODE_LN = 1, MODE_QKV = 2 };

// token index (window layout) -> spatial index (b,z,y,x)
__device__ __forceinline__ int tok2spatial(int t) {
    int widx = t >> 6, n = t & 63;
    int b = widx >> 9, zb = (widx >> 6) & 7, yb = (widx >> 3) & 7, xb = widx & 7;
    int tz = n >> 4, ty = (n >> 2) & 3, tx = n & 3;
    int z = zb * 4 + tz, y = yb * 4 + ty, x = xb * 4 + tx;
    return ((b * 32 + z) * 32 + y) * 32 + x;
}
__device__ __forceinline__ int spatial2tok(int s) {
    int b = s >> 15, z = (s >> 10) & 31, y = (s >> 5) & 31, x = s & 31;
    int widx = ((b * 8 + (z >> 2)) * 8 + (y >> 2)) * 8 + (x >> 2);
    int n = ((z & 3) * 4 + (y & 3)) * 4 + (x & 3);
    return widx * 64 + n;
}
__device__ __forceinline__ float gelu_f(float x) {
    return 0.5f * x * (1.0f + erff(x * 0.70710678118654752f));
}
__device__ __forceinline__ float sigmoid_f(float x) {
    return 1.0f / (1.0f + expf(-x));
}

// CDNA5 async global->LDS copy (16 bytes per active lane), ASYNCcnt-tracked.
__device__ __forceinline__ void async_ld_b128(unsigned lds_off, unsigned gl_off,
                                              const void* sbase) {
    asm volatile("global_load_async_to_lds_b128 %0, %1, %2 offset:0"
                 :
                 : "v"(lds_off), "v"(gl_off), "s"(sbase)
                 : "memory");
}
__device__ __forceinline__ void wait_asynccnt0() {
    asm volatile("s_wait_asynccnt 0x0" ::: "memory");
}

// ---------------------------------------------------------------------------
// Generic fused GEMM: C = A(MxK) @ B(KxN) + bias, with optional fused
// row-LayerNorm or QKV scale/gate epilogue. f32 WMMA 16x16x4.
// Block = 128 threads (4 waves). Each wave owns a 16 x (NT*16) C strip, so a
// wave holds complete output rows -> LayerNorm fuses via half-wave shuffles.
// ---------------------------------------------------------------------------
template <int NT>
__global__ __launch_bounds__(128) void gemm_wmma(
    const float* __restrict__ A, int lda, const float* __restrict__ B, int ldb,
    const float* __restrict__ bias, float* __restrict__ Cout, int ldc, int K,
    int mode, int vol_store, const float* __restrict__ gam,
    const float* __restrict__ bet, const float* __restrict__ gate) {
    const int lane = threadIdx.x & 31;
    const int wv = threadIdx.x >> 5;
    const int m0 = blockIdx.x * 64 + wv * 16;
    const int n0 = blockIdx.y * (NT * 16);
    const int mr = lane & 15;            // row (A/M) or col (B/N) within tile
    const int kh = (lane >> 4) << 1;     // K-pair selector: 0 or 2

    v8f acc[NT];
#pragma unroll
    for (int t = 0; t < NT; ++t)
#pragma unroll
        for (int i = 0; i < 8; ++i) acc[t][i] = 0.0f;

    const float* Ar = A + (size_t)(m0 + mr) * lda;
    for (int k = 0; k < K; k += 4) {
        __builtin_prefetch(Ar + k + 32, 0, 1);
        v2f a;
        a.x = Ar[k + kh];
        a.y = Ar[k + kh + 1];
        const float* Bp = B + (size_t)(k + kh) * ldb + n0 + mr;
#pragma unroll
        for (int t = 0; t < NT; ++t) {
            v2f b;
            b.x = Bp[t * 16];
            b.y = Bp[t * 16 + ldb];
            acc[t] = __builtin_amdgcn_wmma_f32_16x16x4_f32(
                false, a, false, b, (short)0, acc[t], false, false);
        }
    }

    // bias (per column, uniform across the 8 row-slots)
#pragma unroll
    for (int t = 0; t < NT; ++t) {
        float bv = bias[n0 + t * 16 + mr];
#pragma unroll
        for (int r = 0; r < 8; ++r) acc[t][r] += bv;
    }

    if (mode == MODE_LN) {
        const float inv = 1.0f / (float)(NT * 16);
#pragma unroll
        for (int r = 0; r < 8; ++r) {
            float s1 = 0.0f, s2 = 0.0f;
#pragma unroll
            for (int t = 0; t < NT; ++t) {
                float xv = acc[t][r];
                s1 += xv;
                s2 += xv * xv;
            }
            // reduce across the 16 lanes of this half-wave (row elements)
            for (int off = 1; off < 16; off <<= 1) {
                s1 += __shfl_xor(s1, off, 32);
                s2 += __shfl_xor(s2, off, 32);
            }
            float mn = s1 * inv;
            float vr = s2 * inv - mn * mn;
            float rs = rsqrtf(vr + EPSN);
#pragma unroll
            for (int t = 0; t < NT; ++t) {
                int col = n0 + t * 16 + mr;
                acc[t][r] = (acc[t][r] - mn) * rs * gam[col] + bet[col];
            }
        }
    } else if (mode == MODE_QKV) {
        int bidx = (blockIdx.x * 64) >> 15;  // batch of this row-block
#pragma unroll
        for (int t = 0; t < NT; ++t) {
            int col = n0 + t * 16 + mr;
            float mul = 1.0f;
            if (col < 96) mul = 0.25f;                      // q * hd^-0.5
            else if (col >= 192) mul = gate[bidx * 96 + (col - 192)];  // gated v
#pragma unroll
            for (int r = 0; r < 8; ++r) acc[t][r] *= mul;
        }
    }

    const int rbase = m0 + ((lane >> 4) << 3);  // lanes>=16 hold rows M+8
#pragma unroll
    for (int r = 0; r < 8; ++r) {
        int row = rbase + r;
        size_t orow = vol_store ? (size_t)tok2spatial(row) : (size_t)row;
        float* Cr = Cout + orow * (size_t)ldc + n0 + mr;
#pragma unroll
        for (int t = 0; t < NT; ++t) Cr[t * 16] = acc[t][r];
    }
}

// ---------------------------------------------------------------------------
// Depthwise 3x3x3 conv on (b,z,y,x,c) volume + per-(b,c) sum/sumsq stats.
// One block = 192 channels x 64 spatial positions (same batch).
// ---------------------------------------------------------------------------
__global__ __launch_bounds__(192) void dwconv_kernel(
    const float* __restrict__ xcv, const float* __restrict__ dww,
    const float* __restrict__ dwb, float* __restrict__ volc,
    float* __restrict__ stats192) {
    const int c = threadIdx.x;
    const int s0 = blockIdx.x * 64;
    const int b = s0 >> 15;
    float w[27];
#pragma unroll
    for (int i = 0; i < 27; ++i) w[i] = dww[c * 27 + i];
    const float bc = dwb[c];
    float s1 = 0.0f, s2 = 0.0f;
    for (int i = 0; i < 64; ++i) {
        int s = s0 + i;
        int z = (s >> 10) & 31, y = (s >> 5) & 31, x = s & 31;
        float acc = bc;
#pragma unroll
        for (int dz = 0; dz < 3; ++dz) {
            int zz = z + dz - 1;
            if ((unsigned)zz >= 32u) continue;
#pragma unroll
            for (int dy = 0; dy < 3; ++dy) {
                int yy = y + dy - 1;
                if ((unsigned)yy >= 32u) continue;
#pragma unroll
                for (int dx = 0; dx < 3; ++dx) {
                    int xx = x + dx - 1;
                    if ((unsigned)xx >= 32u) continue;
                    acc += w[(dz * 3 + dy) * 3 + dx] *
                           xcv[(size_t)(((b * 32 + zz) * 32 + yy) * 32 + xx) * 192 + c];
                }
            }
        }
        volc[(size_t)s * 192 + c] = acc;
        s1 += acc;
        s2 += acc * acc;
    }
    atomicAdd(&stats192[(b * 192 + c) * 2 + 0], s1);
    atomicAdd(&stats192[(b * 192 + c) * 2 + 1], s2);
}

// InstanceNorm + GELU (in place) + pooled partial sums
__global__ __launch_bounds__(192) void inorm_gelu_pool_kernel(
    float* __restrict__ volc, const float* __restrict__ stats192,
    float* __restrict__ pooledsum) {
    const int c = threadIdx.x;
    const int s0 = blockIdx.x * 64;
    const int b = s0 >> 15;
    const float inv = 1.0f / 32768.0f;
    float mn = stats192[(b * 192 + c) * 2 + 0] * inv;
    float vr = stats192[(b * 192 + c) * 2 + 1] * inv - mn * mn;
    float rs = rsqrtf(vr + EPSN);
    float ps = 0.0f;
    for (int i = 0; i < 64; ++i) {
        size_t idx = (size_t)(s0 + i) * 192 + c;
        float v = gelu_f((volc[idx] - mn) * rs);
        volc[idx] = v;
        ps += v;
    }
    atomicAdd(&pooledsum[b * 192 + c], ps);
}

// channel-interaction MLP: pooled -> gelu(192->24) -> (24->96) -> sigmoid gate
__global__ __launch_bounds__(384) void gate_kernel(
    const float* __restrict__ pooledsum, const float* __restrict__ w1,
    const float* __restrict__ b1, const float* __restrict__ w2,
    const float* __restrict__ b2, float* __restrict__ gate) {
    __shared__ float hid[4 * 24];
    int tid = threadIdx.x;
    if (tid < 96) {
        int b = tid / 24, kk = tid % 24;
        float a = b1[kk];
        for (int c = 0; c < 192; ++c)
            a += (pooledsum[b * 192 + c] * (1.0f / 32768.0f)) * w1[c * 24 + kk];
        hid[b * 24 + kk] = gelu_f(a);
    }
    __syncthreads();
    {
        int b = tid / 96, j = tid % 96;
        float a = b2[j];
        for (int k = 0; k < 24; ++k) a += hid[b * 24 + k] * w2[k * 96 + j];
        gate[b * 96 + j] = sigmoid_f(a);
    }
}

// relative position bias table expansion: (343,6) -> (6,64,64)
__global__ void relbias_kernel(const float* __restrict__ rpb,
                               float* __restrict__ biasA) {
    int i = blockIdx.x * 256 + threadIdx.x;
    if (i >= 6 * 64 * 64) return;
    int h = i >> 12;
    int rem = i & 4095;
    int n = rem >> 6, m = rem & 63;
    int nz = n >> 4, ny = (n >> 2) & 3, nx = n & 3;
    int mz = m >> 4, my = (m >> 2) & 3, mx = m & 3;
    int idx = (nz - mz + 3) * 49 + (ny - my + 3) * 7 + (nx - mx + 3);
    biasA[i] = rpb[idx * 6 + h];
}

// ---------------------------------------------------------------------------
// Window attention. One block = 2 windows; 8 waves = 2 windows x 4 M-tiles.
// Loop over 6 heads: K/V for both windows are staged into LDS with CDNA5
// async global->LDS b128 copies (ASYNCcnt + s_wait_asynccnt 0), then QK^T and
// P@V run on f32 WMMA 16x16x4. Softmax uses half-wave shuffles; P round-trips
// through a private LDS strip to switch from C-layout to A-layout.
// ---------------------------------------------------------------------------
__global__ __launch_bounds__(256) void attn_kernel(
    const float* __restrict__ qkv, const float* __restrict__ biasA,
    float* __restrict__ xa2) {
    __shared__ float kvbuf[2 * 64 * 2 * 16];  // [win][tok][k/v][d]  16 KB
    __shared__ float pstrip[8 * 1024];        // per-wave P strips   32 KB
    const int lane = threadIdx.x & 31;
    const int wv = threadIdx.x >> 5;
    const int winl = wv >> 2;                 // local window 0/1
    const int widx = blockIdx.x * 2 + winl;
    const int mt = wv & 3;
    const int mr = lane & 15;
    const int kh = (lane >> 4) << 1;
    const int hshift = (lane >> 4) << 3;      // +8 row offset, upper half-wave
    float* P = pstrip + wv * 1024;
    const float* qbase = qkv + (size_t)widx * 64 * 288;

    // this thread's staging segment: 16 consecutive floats of one (win,tok,k/v)
    const int seg = threadIdx.x;              // 0..255
    const int sw = seg >> 7, srem = seg & 127;
    const int stok = srem >> 1, skv = srem & 1;
    const unsigned lbase =
        (unsigned)(uintptr_t)&kvbuf[((sw * 64 + stok) * 2 + skv) * 16];
    const unsigned gbase0 =
        (unsigned)((((unsigned)(blockIdx.x * 2 + sw) * 64 + stok) * 288 + 96 +
                    skv * 96) * 4u);

    for (int head = 0; head < 6; ++head) {
        // ---- async stage K,V for this head (both windows) ----
        unsigned gb = gbase0 + (unsigned)(head * 16 * 4);
#pragma unroll
        for (int j = 0; j < 4; ++j)
            async_ld_b128(lbase + j * 16, gb + j * 16, (const void*)qkv);
        wait_asynccnt0();
        __syncthreads();

        const int qo = head * 16;
        const float* kloc = &kvbuf[(winl * 64) * 2 * 16];  // k at +0, v at +16

        // scores = q @ k^T   (q pre-scaled by hd^-0.5 in the QKV GEMM)
        v8f sc[4];
#pragma unroll
        for (int nt = 0; nt < 4; ++nt)
#pragma unroll
            for (int i = 0; i < 8; ++i) sc[nt][i] = 0.0f;
#pragma unroll
        for (int kk = 0; kk < 4; ++kk) {
            v2f a;
            a.x = qbase[(mt * 16 + mr) * 288 + qo + kk * 4 + kh];
            a.y = qbase[(mt * 16 + mr) * 288 + qo + kk * 4 + kh + 1];
#pragma unroll
            for (int nt = 0; nt < 4; ++nt) {
                v2f b;
                b.x = kloc[(nt * 16 + mr) * 32 + kk * 4 + kh];
                b.y = kloc[(nt * 16 + mr) * 32 + kk * 4 + kh + 1];
                sc[nt] = __builtin_amdgcn_wmma_f32_16x16x4_f32(
                    false, a, false, b, (short)0, sc[nt], false, false);
            }
        }

        // + relative position bias, row softmax, write P to LDS strip
#pragma unroll
        for (int r = 0; r < 8; ++r) {
            int rowl = mt * 16 + r + hshift;
            float e0, e1, e2, e3, mx;
            e0 = sc[0][r] + biasA[(head * 64 + rowl) * 64 + 0 * 16 + mr];
            e1 = sc[1][r] + biasA[(head * 64 + rowl) * 64 + 1 * 16 + mr];
            e2 = sc[2][r] + biasA[(head * 64 + rowl) * 64 + 2 * 16 + mr];
            e3 = sc[3][r] + biasA[(head * 64 + rowl) * 64 + 3 * 16 + mr];
            mx = fmaxf(fmaxf(e0, e1), fmaxf(e2, e3));
            for (int off = 1; off < 16; off <<= 1)
                mx = fmaxf(mx, __shfl_xor(mx, off, 32));
            e0 = expf(e0 - mx); e1 = expf(e1 - mx);
            e2 = expf(e2 - mx); e3 = expf(e3 - mx);
            float sm = e0 + e1 + e2 + e3;
            for (int off = 1; off < 16; off <<= 1) sm += __shfl_xor(sm, off, 32);
            float rinv = 1.0f / sm;
            int pr = (r + hshift) * 64;
            P[pr + 0 * 16 + mr] = e0 * rinv;
            P[pr + 1 * 16 + mr] = e1 * rinv;
            P[pr + 2 * 16 + mr] = e2 * rinv;
            P[pr + 3 * 16 + mr] = e3 * rinv;
        }
        __syncthreads();

        // out = P @ V  (K = 64 key tokens, V from LDS)
        v8f o;
#pragma unroll
        for (int i = 0; i < 8; ++i) o[i] = 0.0f;
#pragma unroll
        for (int ks = 0; ks < 16; ++ks) {
            v2f a;
            a.x = P[mr * 64 + ks * 4 + kh];
            a.y = P[mr * 64 + ks * 4 + kh + 1];
            v2f b;
            b.x = kloc[(ks * 4 + kh) * 32 + 16 + mr];
            b.y = kloc[(ks * 4 + kh + 1) * 32 + 16 + mr];
            o = __builtin_amdgcn_wmma_f32_16x16x4_f32(
                false, a, false, b, (short)0, o, false, false);
        }
#pragma unroll
        for (int r = 0; r < 8; ++r)
            xa2[(size_t)(widx * 64 + mt * 16 + r + hshift) * 96 + qo + mr] = o[r];
        __syncthreads();  // all waves done with kvbuf before next head restage
    }
}

// LayerNorm over 96 channels per token -> xcat[:, 0:96]
__global__ __launch_bounds__(256) void ln96_kernel(
    const float* __restrict__ xa2, const float* __restrict__ g,
    const float* __restrict__ be, float* __restrict__ xcat) {
    int lane = threadIdx.x & 31;
    int wv = threadIdx.x >> 5;
    int t = blockIdx.x * 8 + wv;
    const float* row = xa2 + (size_t)t * 96;
    float e0 = row[lane], e1 = row[lane + 32], e2 = row[lane + 64];
    float s1 = e0 + e1 + e2;
    float s2 = e0 * e0 + e1 * e1 + e2 * e2;
    for (int off = 1; off < 32; off <<= 1) {
        s1 += __shfl_xor(s1, off, 32);
        s2 += __shfl_xor(s2, off, 32);
    }
    float mn = s1 * (1.0f / 96.0f);
    float vr = s2 * (1.0f / 96.0f) - mn * mn;
    float rs = rsqrtf(vr + EPSN);
    float* o = xcat + (size_t)t * 192;
    o[lane]      = (e0 - mn) * rs * g[lane]      + be[lane];
    o[lane + 32] = (e1 - mn) * rs * g[lane + 32] + be[lane + 32];
    o[lane + 64] = (e2 - mn) * rs * g[lane + 64] + be[lane + 64];
}

// spatial-interaction conv1: 96 -> 12 per voxel + per-(b,j) stats
__global__ __launch_bounds__(128) void si1_kernel(
    const float* __restrict__ xa2, const float* __restrict__ w1,
    const float* __restrict__ b1, float* __restrict__ t12,
    float* __restrict__ stats12) {
    int s = blockIdx.x * 128 + threadIdx.x;
    int lane = threadIdx.x & 31;
    int b = s >> 15;
    int t = spatial2tok(s);
    const float* row = xa2 + (size_t)t * 96;
    float acc[12];
#pragma unroll
    for (int j = 0; j < 12; ++j) acc[j] = b1[j];
    for (int c = 0; c < 96; ++c) {
        float xv = row[c];
#pragma unroll
        for (int j = 0; j < 12; ++j) acc[j] += xv * w1[c * 12 + j];
    }
#pragma unroll
    for (int j = 0; j < 12; ++j) t12[(size_t)s * 12 + j] = acc[j];
#pragma unroll
    for (int j = 0; j < 12; ++j) {
        float r1 = acc[j], r2 = acc[j] * acc[j];
        for (int off = 1; off < 32; off <<= 1) {
            r1 += __shfl_xor(r1, off, 32);
            r2 += __shfl_xor(r2, off, 32);
        }
        if (lane == 0) {
            atomicAdd(&stats12[(b * 12 + j) * 2 + 0], r1);
            atomicAdd(&stats12[(b * 12 + j) * 2 + 1], r2);
        }
    }
}

// inorm+gelu over t12, conv2 (12->1), sigmoid gate applied to pvol (in place),
// plus per-(b,c) stats for the final InstanceNorm.
__global__ __launch_bounds__(96) void si2_kernel(
    const float* __restrict__ t12, const float* __restrict__ stats12,
    const float* __restrict__ w2, const float* __restrict__ b2,
    float* __restrict__ pvol, float* __restrict__ stats96) {
    int c = threadIdx.x;  // 0..95
    int s0 = blockIdx.x * 64;
    int b = s0 >> 15;
    float m12[12], r12[12], w2r[12];
    const float inv = 1.0f / 32768.0f;
#pragma unroll
    for (int j = 0; j < 12; ++j) {
        float mn = stats12[(b * 12 + j) * 2 + 0] * inv;
        float vr = stats12[(b * 12 + j) * 2 + 1] * inv - mn * mn;
        m12[j] = mn;
        r12[j] = rsqrtf(vr + EPSN);
        w2r[j] = w2[j];
    }
    float bb = b2[0];
    float s1 = 0.0f, s2 = 0.0f;
    for (int i = 0; i < 64; ++i) {
        int s = s0 + i;
        float a = bb;
#pragma unroll
        for (int j = 0; j < 12; ++j)
            a += gelu_f((t12[(size_t)s * 12 + j] - m12[j]) * r12[j]) * w2r[j];
        float g = sigmoid_f(a);
        size_t idx = (size_t)s * 96 + c;
        float p = g * pvol[idx];
        pvol[idx] = p;
        s1 += p;
        s2 += p * p;
    }
    atomicAdd(&stats96[(b * 96 + c) * 2 + 0], s1);
    atomicAdd(&stats96[(b * 96 + c) * 2 + 1], s2);
}

// final InstanceNorm of the gated CNN branch -> xcat[:, 96:192] (window layout)
__global__ __launch_bounds__(256) void si3_kernel(
    const float* __restrict__ pvol, const float* __restrict__ stats96,
    float* __restrict__ xcat) {
    int i = blockIdx.x * 256 + threadIdx.x;  // over 131072*96
    int s = i / 96, c = i - s * 96;
    int b = s >> 15;
    const float inv = 1.0f / 32768.0f;
    float mn = stats96[(b * 96 + c) * 2 + 0] * inv;
    float vr = stats96[(b * 96 + c) * 2 + 1] * inv - mn * mn;
    float rs = rsqrtf(vr + EPSN);
    int t = spatial2tok(s);
    xcat[(size_t)t * 192 + 96 + c] = (pvol[i] - mn) * rs;
}

__global__ void zero_kernel(float* __restrict__ p, int n) {
    int i = blockIdx.x * 256 + threadIdx.x;
    if (i < n) p[i] = 0.0f;
}

// ---------------------------------------------------------------------------
// Orchestration. Fixed shapes from setup_inputs(): Bwin=2048, N=64, C=192,
// D=H=W=32, b=4, spatial=32768/batch, M=131072 token rows.
// ---------------------------------------------------------------------------
extern "C" void kernel_launch(void* const* d_in, const int* in_sizes, int n_in,
                              void* d_out, int out_size, void* d_ws,
                              size_t ws_size, hipStream_t stream) {
    (void)in_sizes; (void)n_in; (void)out_size; (void)ws_size;
    const float* x       = (const float*)d_in[0];
    const float* w_pattn = (const float*)d_in[1];
    const float* b_pattn = (const float*)d_in[2];
    const float* g_aln   = (const float*)d_in[3];
    const float* b_aln   = (const float*)d_in[4];
    const float* w_pcnn  = (const float*)d_in[5];
    const float* b_pcnn  = (const float*)d_in[6];
    const float* g_cln   = (const float*)d_in[7];
    const float* b_cln   = (const float*)d_in[8];
    const float* dw_w    = (const float*)d_in[9];
    const float* dw_b    = (const float*)d_in[10];
    const float* ci_w1   = (const float*)d_in[11];
    const float* ci_b1   = (const float*)d_in[12];
    const float* ci_w2   = (const float*)d_in[13];
    const float* ci_b2   = (const float*)d_in[14];
    const float* pj_w    = (const float*)d_in[15];
    const float* pj_b    = (const float*)d_in[16];
    const float* qkv_w   = (const float*)d_in[17];
    const float* qkv_b   = (const float*)d_in[18];
    const float* si_w1   = (const float*)d_in[19];
    const float* si_b1   = (const float*)d_in[20];
    const float* si_w2   = (const float*)d_in[21];
    const float* si_b2   = (const float*)d_in[22];
    const float* g_anorm = (const float*)d_in[23];
    const float* b_anorm = (const float*)d_in[24];
    const float* w_proj  = (const float*)d_in[25];
    const float* b_proj  = (const float*)d_in[26];
    const float* rpb     = (const float*)d_in[27];

    float* ws = (float*)d_ws;
    float* XA    = ws;                 // 131072x96   LN'd attention input
    float* XCV   = ws + 12582912;      // (b,z,y,x,192) LN'd CNN input
    float* VOLC  = ws + 37748736;      // conv out -> gelu(inorm) in place
    float* PVOL  = ws + 62914560;      // 131072x96   1x1-proj volume
    float* QKVB  = ws + 75497472;      // 131072x288
    float* XA2   = ws + 113246208;     // 131072x96   attention output
    float* T12   = ws + 125829120;     // 131072x12
    float* XCAT  = ws + 127401984;     // 131072x192  concat buffer
    float* STATS = ws + 152567808;     // small stats region (zeroed each call)
    float* S192  = STATS;              // 4*192*2
    float* PSUM  = STATS + 1536;       // 4*192
    float* S12   = STATS + 2304;       // 4*12*2
    float* S96   = STATS + 2400;       // 4*96*2
    float* GATE  = STATS + 3168;       // 4*96
    float* BIASA = STATS + 3552;       // 6*64*64

    zero_kernel<<<(3168 + 255) / 256, 256, 0, stream>>>(STATS, 3168);

    // x_atten = LN(x @ w_pattn + b)   (N=96, fused LN)
    gemm_wmma<6><<<dim3(2048, 1), 128, 0, stream>>>(
        x, 192, w_pattn, 96, b_pattn, XA, 96, 192, MODE_LN, 0, g_aln, b_aln, nullptr);
    // x_cnn = LN(x @ w_pcnn + b), scattered into (b,z,y,x,c) volume layout
    gemm_wmma<12><<<dim3(2048, 1), 128, 0, stream>>>(
        x, 192, w_pcnn, 192, b_pcnn, XCV, 192, 192, MODE_LN, 1, g_cln, b_cln, nullptr);

    dwconv_kernel<<<2048, 192, 0, stream>>>(XCV, dw_w, dw_b, VOLC, S192);
    inorm_gelu_pool_kernel<<<2048, 192, 0, stream>>>(VOLC, S192, PSUM);
    gate_kernel<<<1, 384, 0, stream>>>(PSUM, ci_w1, ci_b1, ci_w2, ci_b2, GATE);
    relbias_kernel<<<96, 256, 0, stream>>>(rpb, BIASA);

    // 1x1 conv 192->96 on gelu volume
    gemm_wmma<6><<<dim3(2048, 1), 128, 0, stream>>>(
        VOLC, 192, pj_w, 96, pj_b, PVOL, 96, 192, MODE_PLAIN, 0, nullptr, nullptr, nullptr);
    // QKV with fused q-scale and v-gate (N=288 in 3 column chunks)
    gemm_wmma<6><<<dim3(2048, 3), 128, 0, stream>>>(
        XA, 96, qkv_w, 288, qkv_b, QKVB, 288, 96, MODE_QKV, 0, nullptr, nullptr, GATE);

    attn_kernel<<<1024, 256, 0, stream>>>(QKVB, BIASA, XA2);

    ln96_kernel<<<16384, 256, 0, stream>>>(XA2, g_anorm, b_anorm, XCAT);
    si1_kernel<<<1024, 128, 0, stream>>>(XA2, si_w1, si_b1, T12, S12);
    si2_kernel<<<2048, 96, 0, stream>>>(T12, S12, si_w2, si_b2, PVOL, S96);
    si3_kernel<<<49152, 256, 0, stream>>>(PVOL, S96, XCAT);

    // out = concat(LN(attn), cnn) @ w_proj + b  (N=192 in 2 column chunks)
    gemm_wmma<6><<<dim3(2048, 2), 128, 0, stream>>>(
        XCAT, 192, w_proj, 192, b_proj, (float*)d_out, 192, 192, MODE_PLAIN, 0,
        nullptr, nullptr, nullptr);
}